// MultiBandedROIAlign_60181081752145
// MI455X (gfx1250) — compile-verified
//
#include <hip/hip_runtime.h>

#define OUTSZ 7
#define SR 2
#define SGRID (OUTSZ*SR)        // 14
#define NSAMP (SGRID*SGRID)     // 196
#define NPIX (OUTSZ*OUTSZ)      // 49
#define NCH 256
#define FH 200
#define FW 200
#define BH (FH/2)               // 100 (band height)
#define CHW (FH*FW)             // 40000
#define CH_GROUP 16

typedef __attribute__((ext_vector_type(2))) float v2f;
typedef __attribute__((ext_vector_type(8))) float v8f;

__device__ __forceinline__ v8f pool_step(v8f acc, const float* sval,
                                         int m, int kh, int p, int kp)
{
    // A: lane m = channel row; VGPR0 -> K=2*kh, VGPR1 -> K=2*kh+1
    const v2f a = *(const v2f*)&sval[m*NSAMP + p*4 + kh*2];
    // B: 0.25 in output-pixel column kp (all 4 K rows identical)
    const float bw = (m == kp) ? 0.25f : 0.0f;
    v2f b; b.x = bw; b.y = bw;
    return __builtin_amdgcn_wmma_f32_16x16x4_f32(
               false, a, false, b, (short)0, acc, false, false);
}

__global__ __launch_bounds__(256)
void roi_align_wmma(const float* __restrict__ feat,
                    const float* __restrict__ boxes,
                    float* __restrict__ out, int K)
{
    // val[16 local channels][196 samples], samples stored pixel-major:
    // pos = pixel*4 + (sy*2+sx)  ->  pooling matrix is block-diagonal.
    __shared__ float sval[CH_GROUP * NSAMP];

    const int bid  = blockIdx.x;
    const int kbox = bid >> 1;
    const int band = bid & 1;

    // ---- box parameters (wave-uniform broadcast loads) ----
    const float bidxf = boxes[kbox*5 + 0];
    const float x1    = boxes[kbox*5 + 1];
    const float y1    = boxes[kbox*5 + 2];
    const float x2    = boxes[kbox*5 + 3];
    const float y2    = boxes[kbox*5 + 4];
    const int   n0    = (int)bidxf;
    const float roi_w = fmaxf(x2 - x1, 1.0f);
    const float roi_h = fmaxf(y2 - y1, 1.0f);
    const float bin_w = roi_w / (float)OUTSZ;
    const float bin_h = roi_h / (float)OUTSZ;

    const int t = threadIdx.x;

    // ---- per-sample geometry: computed ONCE, reused for all 256 channels ----
    int   o00=0, o01=0, o10=0, o11=0, pos=0;
    float w00=0.f, w01=0.f, w10=0.f, w11=0.f;
    const bool sampler = (t < NSAMP);
    if (sampler) {
        const int iy = t / SGRID;
        const int ix = t - iy*SGRID;
        const int py = iy >> 1, sy = iy & 1;
        const int px = ix >> 1, sx = ix & 1;
        pos = (py*OUTSZ + px)*4 + (sy*2 + sx);

        float y = y1 + ((float)iy*0.5f + 0.25f) * bin_h;
        float x = x1 + ((float)ix*0.5f + 0.25f) * bin_w;
        const bool valid = (y >= -1.0f) && (y <= (float)BH) &&
                           (x >= -1.0f) && (x <= (float)FW);
        y = fminf(fmaxf(y, 0.0f), (float)(BH-1));
        x = fminf(fmaxf(x, 0.0f), (float)(FW-1));
        const int ylo = (int)floorf(y);
        const int xlo = (int)floorf(x);
        const int yhi = min(ylo + 1, BH-1);
        const int xhi = min(xlo + 1, FW-1);
        const float ly = y - (float)ylo, lx = x - (float)xlo;
        const float hy = 1.0f - ly,      hx = 1.0f - lx;
        w00 = hy*hx; w01 = hy*lx; w10 = ly*hx; w11 = ly*lx;
        if (!valid) { w00 = w01 = w10 = w11 = 0.0f; }
        const int rowlo = (band*BH + ylo)*FW;
        const int rowhi = (band*BH + yhi)*FW;
        o00 = rowlo + xlo; o01 = rowlo + xhi;
        o10 = rowhi + xlo; o11 = rowhi + xhi;
    }

    const size_t imgBase = (size_t)n0 * (size_t)NCH * (size_t)CHW;
    const int lane = t & 31;
    // SGPR-uniform wave id: keeps the WMMA region in provably uniform control
    // flow (EXEC all-ones, scalar loop bounds, no per-lane exec bookkeeping).
    const int wv   = __builtin_amdgcn_readfirstlane(t >> 5);
    const int m    = lane & 15;   // A: channel row.  C/D & stores: pixel col N.
    const int kh   = lane >> 4;   // A: K-pair half.  C/D: M half (+8 rows).

    const size_t outBase = (size_t)kbox * (size_t)(2*NCH*NPIX)
                         + (size_t)(band*NCH) * (size_t)NPIX;

    for (int c0 = 0; c0 < NCH; c0 += CH_GROUP) {
        // phase 1: gather + bilinear for 16 channels (L2-resident gathers)
        if (sampler) {
            #pragma unroll 4
            for (int cu = 0; cu < CH_GROUP; ++cu) {
                const float* fc = feat + imgBase + (size_t)(c0 + cu)*CHW;
                const float v = w00*fc[o00] + w01*fc[o01]
                              + w10*fc[o10] + w11*fc[o11];
                sval[cu*NSAMP + pos] = v;
            }
        }
        __syncthreads();

        // phase 2: 2x2 average-pool as WMMA:
        //   out[16c x 16p] += A[16c x 4samp] * B[4samp x 16p]
        // Block-diagonal pooling matrix -> exactly 1 k-step per output pixel.
        if (wv < 4) {
            v8f acc = {};
            if (wv < 3) {
                const int pbase = wv * 16;
                #pragma unroll
                for (int kp = 0; kp < 16; ++kp)
                    acc = pool_step(acc, sval, m, kh, pbase + kp, kp);
            } else {
                acc = pool_step(acc, sval, m, kh, 48, 0);  // last pixel tile
            }
            // C/D: VGPR r, lanes0-15: M=r, lanes16-31: M=8+r; N=lane&15
            const int p = wv*16 + m;
            if (p < NPIX) {
                float* op = out + outBase + (size_t)(c0 + 8*kh)*NPIX + p;
                #pragma unroll
                for (int r = 0; r < 8; ++r)
                    op[r*NPIX] = acc[r];
            }
        }
        __syncthreads();
    }
}

extern "C" void kernel_launch(void* const* d_in, const int* in_sizes, int n_in,
                              void* d_out, int out_size, void* d_ws, size_t ws_size,
                              hipStream_t stream)
{
    const float* feat  = (const float*)d_in[0];
    const float* boxes = (const float*)d_in[1];
    float*       out   = (float*)d_out;
    const int K = in_sizes[1] / 5;   // 512 boxes
    roi_align_wmma<<<dim3(2*K), dim3(256), 0, stream>>>(feat, boxes, out, K);
}